// GATLayer_53936199303340
// MI455X (gfx1250) — compile-verified
//
#include <hip/hip_runtime.h>

typedef float v2f __attribute__((ext_vector_type(2)));
typedef float v8f __attribute__((ext_vector_type(8)));
typedef unsigned int u32x4 __attribute__((ext_vector_type(4)));
typedef int i32x4 __attribute__((ext_vector_type(4)));
typedef int i32x8 __attribute__((ext_vector_type(8)));

#define HEADS 4
#define CPH 32          // channels per head
#define HC 128          // HEADS * CPH
#define INFEAT 128      // input features
#define NEG_SLOPE 0.2f

// LDS row strides (DWORDs) chosen for bank-conflict-free WMMA operand reads
#define W_LDS_STRIDE 80   // 64 tile cols + 16 pad  (2 rows apart -> +32 banks)
#define A_LDS_STRIDE 132  // 128 tile cols + 4 pad  (rows stride 4 banks apart)

__device__ __forceinline__ float lrelu(float v) { return v > 0.f ? v : v * NEG_SLOPE; }

// Monotonic float<->uint encoding so float max == unsigned max
__device__ __forceinline__ unsigned fenc(float f) {
  unsigned u = __float_as_uint(f);
  return (u & 0x80000000u) ? ~u : (u | 0x80000000u);
}
__device__ __forceinline__ float fdec(unsigned e) {
  unsigned u = (e & 0x80000000u) ? (e ^ 0x80000000u) : ~e;
  return __uint_as_float(u);
}

// ---------------------------------------------------------------------------
// Tensor Data Mover: async 2D tile load Global -> LDS (CDNA5 ISA ch. 7/8).
// D# group0: count=1 | lds_addr | global_addr | type=2.
// D# group1: data_size=4B, pad_enable + interval/amount codes, tensor dims,
//            tile dims, dim0 stride. Groups 2/3 zero (2D tile, tile_dim2=0).
// ---------------------------------------------------------------------------
__device__ __forceinline__ void tdm_load_2d(unsigned lds_off, const void* gptr,
                                            unsigned tensor_d0, unsigned tensor_d1,
                                            unsigned tile_d0, unsigned tile_d1,
                                            unsigned stride0,
                                            unsigned pad_interval_code,
                                            unsigned pad_amount_code) {
  unsigned long long ga = (unsigned long long)(size_t)gptr;
  u32x4 g0;
  g0[0] = 1u;                                            // count=1, user desc
  g0[1] = lds_off;                                       // LDS byte address
  g0[2] = (unsigned)(ga & 0xFFFFFFFFu);                  // global_addr[31:0]
  g0[3] = (unsigned)((ga >> 32) & 0x01FFFFFFu) | (2u << 30);  // [56:32] | type=2
  i32x8 g1;
  g1[0] = (int)((2u << 16)                               // data_size = 4 bytes
                | (1u << 20)                             // pad_enable
                | (pad_interval_code << 22)
                | (pad_amount_code << 25));
  g1[1] = (int)((tensor_d0 & 0xFFFFu) << 16);            // tensor_dim0 lo16 @ bit48
  g1[2] = (int)(((tensor_d0 >> 16) & 0xFFFFu) | ((tensor_d1 & 0xFFFFu) << 16));
  g1[3] = (int)(((tensor_d1 >> 16) & 0xFFFFu) | ((tile_d0 & 0xFFFFu) << 16));
  g1[4] = (int)(tile_d1 & 0xFFFFu);                      // tile_dim1, tile_dim2=0
  g1[5] = (int)stride0;                                  // tensor_dim0_stride lo32
  g1[6] = 0;
  g1[7] = 0;
  i32x4 z4 = {0, 0, 0, 0};
#if __clang_major__ >= 23
  i32x8 z8 = {0, 0, 0, 0, 0, 0, 0, 0};
  __builtin_amdgcn_tensor_load_to_lds(g0, g1, z4, z4, z8, 0);
#else
  __builtin_amdgcn_tensor_load_to_lds(g0, g1, z4, z4, 0);
#endif
}

// ---------------------------------------------------------------------------
// 1) h = x @ W   [n,128] x [128,128] -> [n,128], fp32 WMMA 16x16x4.
//    Block = 4 waves. blockIdx.x = M-tile (16 rows), blockIdx.y selects a
//    64-column slab of W; threadIdx.y picks the 16-col N-tile in the slab.
//    TDM stages the W slab (128x64, padded rows) and the A tile (16x128,
//    padded rows) into LDS; WMMA operands then come from LDS.
// ---------------------------------------------------------------------------
__global__ void gat_gemm_wmma(const float* __restrict__ x,
                              const float* __restrict__ W,
                              float* __restrict__ h, int n) {
  __shared__ float ldsW[128 * W_LDS_STRIDE];  // 40 KB
  __shared__ float ldsA[16 * A_LDS_STRIDE];   //  8.25 KB

  const int lane = threadIdx.x;        // 0..31
  const int half = lane >> 4;          // 0: lanes 0-15, 1: lanes 16-31
  const int lid  = lane & 15;
  const int row0 = blockIdx.x * 16;    // M-tile base
  const int cbase = blockIdx.y * 64;   // W slab base column
  const int col0 = cbase + threadIdx.y * 16;
  const int wcol = threadIdx.y * 16 + lid;  // column within the LDS slab

  if (threadIdx.y == 0) {
    // W slab: tensor row length 128, 128 rows; tile = 64 cols x 128 rows.
    // pad: every 64 DWORDs (code 5) add 16 DWORDs (code 15) -> row stride 80.
    tdm_load_2d((unsigned)(size_t)ldsW, W + cbase,
                /*tensor_d0=*/INFEAT, /*tensor_d1=*/INFEAT,
                /*tile_d0=*/64, /*tile_d1=*/128, /*stride0=*/HC,
                /*interval=*/5u, /*amount=*/15u);
    // A tile: 16 rows of x starting at row0; tile = 128 cols x 16 rows.
    // pad: every 128 DWORDs (code 6) add 4 DWORDs (code 3) -> row stride 132.
    unsigned rows_left = (unsigned)(n - row0);
    tdm_load_2d((unsigned)(size_t)ldsA, x + (size_t)row0 * INFEAT,
                /*tensor_d0=*/INFEAT, /*tensor_d1=*/rows_left,
                /*tile_d0=*/INFEAT, /*tile_d1=*/16, /*stride0=*/INFEAT,
                /*interval=*/6u, /*amount=*/3u);
    __builtin_amdgcn_s_wait_tensorcnt(0);
  }
  __syncthreads();

  v8f acc = {};
#pragma unroll
  for (int k0 = 0; k0 < INFEAT; k0 += 4) {
    const int ka = k0 + half * 2;      // A: lanes 16-31 hold K+2 (ISA 7.12.2)
    v2f a = *(const v2f*)(ldsA + lid * A_LDS_STRIDE + ka);
    v2f b;                             // B 4x16: VGPR0 rows {0,2}, VGPR1 rows {1,3}
    b.x = ldsW[(ka + 0) * W_LDS_STRIDE + wcol];
    b.y = ldsW[(ka + 1) * W_LDS_STRIDE + wcol];
    acc = __builtin_amdgcn_wmma_f32_16x16x4_f32(false, a, false, b,
                                                (short)0, acc, false, false);
  }
#pragma unroll
  for (int v = 0; v < 8; ++v) {        // D: VGPR v -> M = v (+8 for lanes 16-31)
    const int row = row0 + v + half * 8;
    if (row < n) h[(size_t)row * HC + col0 + lid] = acc[v];
  }
}

// ---------------------------------------------------------------------------
// 2) alpha_src / alpha_dst : per (node, head) dot with a_src / a_dst
// ---------------------------------------------------------------------------
__global__ void gat_alpha(const float* __restrict__ h,
                          const float* __restrict__ a_src,
                          const float* __restrict__ a_dst,
                          float* __restrict__ as, float* __restrict__ ad, int nh) {
  int t = blockIdx.x * blockDim.x + threadIdx.x;
  if (t >= nh) return;
  int node = t >> 2, head = t & 3;
  const float* hp = h + (size_t)node * HC + head * CPH;
  const float* sp = a_src + head * CPH;
  const float* dp = a_dst + head * CPH;
  float sa = 0.f, da = 0.f;
#pragma unroll
  for (int c = 0; c < CPH; ++c) {
    float v = hp[c];
    sa += v * sp[c];
    da += v * dp[c];
  }
  as[t] = sa;
  ad[t] = da;
}

// 3) seed segment-max with the self-loop logit (add_self_loops=True)
__global__ void gat_init_max(const float* __restrict__ as, const float* __restrict__ ad,
                             unsigned* __restrict__ menc, int nh) {
  int t = blockIdx.x * blockDim.x + threadIdx.x;
  if (t >= nh) return;
  menc[t] = fenc(lrelu(as[t] + ad[t]));
}

// 4) per-edge atomic max of the logit into the destination node
__global__ void gat_edge_max(const int* __restrict__ ei,
                             const float* __restrict__ as, const float* __restrict__ ad,
                             unsigned* __restrict__ menc, int E) {
  int t = blockIdx.x * blockDim.x + threadIdx.x;
  if (t >= E * HEADS) return;
  int e = t >> 2, head = t & 3;
  int src = ei[e], dst = ei[E + e];
  float v = lrelu(as[src * HEADS + head] + ad[dst * HEADS + head]);
  atomicMax(&menc[dst * HEADS + head], fenc(v));
}

// 5) decode max, seed denom with the self-loop exp term
__global__ void gat_init_denom(const float* __restrict__ as, const float* __restrict__ ad,
                               const unsigned* __restrict__ menc,
                               float* __restrict__ mf, float* __restrict__ denom, int nh) {
  int t = blockIdx.x * blockDim.x + threadIdx.x;
  if (t >= nh) return;
  float m = fdec(menc[t]);
  mf[t] = m;
  denom[t] = __expf(lrelu(as[t] + ad[t]) - m);
}

// 6) per-edge atomic add of exp(logit - max) into denom
__global__ void gat_edge_denom(const int* __restrict__ ei,
                               const float* __restrict__ as, const float* __restrict__ ad,
                               const float* __restrict__ mf, float* __restrict__ denom, int E) {
  int t = blockIdx.x * blockDim.x + threadIdx.x;
  if (t >= E * HEADS) return;
  int e = t >> 2, head = t & 3;
  int src = ei[e], dst = ei[E + e];
  int di = dst * HEADS + head;
  float w = __expf(lrelu(as[src * HEADS + head] + ad[di]) - mf[di]);
  atomicAdd(&denom[di], w);
}

// 7) out = bias + self-loop attention * h[node]  (full overwrite of d_out)
__global__ void gat_init_out(const float* __restrict__ h,
                             const float* __restrict__ as, const float* __restrict__ ad,
                             const float* __restrict__ mf, const float* __restrict__ denom,
                             const float* __restrict__ bias,
                             float* __restrict__ out, int n) {
  int t = blockIdx.x * blockDim.x + threadIdx.x;
  if (t >= n * HC) return;
  int node = t >> 7, hc = t & 127, head = hc >> 5;
  int ni = node * HEADS + head;
  float el = lrelu(as[ni] + ad[ni]);
  float att = __expf(el - mf[ni]) / denom[ni];
  out[t] = bias[hc] + att * h[t];
}

// 8) one wave per edge: scatter att * h[src] into out[dst] (128 channels)
__global__ void gat_edge_aggr(const int* __restrict__ ei, const float* __restrict__ h,
                              const float* __restrict__ as, const float* __restrict__ ad,
                              const float* __restrict__ mf, const float* __restrict__ denom,
                              float* __restrict__ out, int E) {
  int gid = blockIdx.x * blockDim.x + threadIdx.x;
  int wid = gid >> 5;        // one wave32 per edge
  int lane = gid & 31;
  if (wid >= E) return;
  int src = ei[wid], dst = ei[E + wid];
  float att = 0.f;
  if (lane < HEADS) {
    int di = dst * HEADS + lane;
    float e = lrelu(as[src * HEADS + lane] + ad[di]);
    att = __expf(e - mf[di]) / denom[di];
  }
  const float* hs = h + (size_t)src * HC;
  float* od = out + (size_t)dst * HC;
#pragma unroll
  for (int head = 0; head < HEADS; ++head) {
    float a = __shfl(att, head, 32);
    int c = head * CPH + lane;
    atomicAdd(&od[c], a * hs[c]);
  }
}

// ---------------------------------------------------------------------------
extern "C" void kernel_launch(void* const* d_in, const int* in_sizes, int n_in,
                              void* d_out, int out_size, void* d_ws, size_t ws_size,
                              hipStream_t stream) {
  const float* x     = (const float*)d_in[0];
  const int*   ei    = (const int*)d_in[1];
  const float* W     = (const float*)d_in[2];
  const float* a_src = (const float*)d_in[3];
  const float* a_dst = (const float*)d_in[4];
  const float* bias  = (const float*)d_in[5];
  float* out = (float*)d_out;

  const int n  = in_sizes[0] / INFEAT;
  const int E  = in_sizes[1] / 2;
  const int nh = n * HEADS;

  // workspace carve-up
  char* ws = (char*)d_ws;
  float*    h     = (float*)ws;    ws += (size_t)n * HC * sizeof(float);
  float*    as    = (float*)ws;    ws += (size_t)nh * sizeof(float);
  float*    ad    = (float*)ws;    ws += (size_t)nh * sizeof(float);
  unsigned* menc  = (unsigned*)ws; ws += (size_t)nh * sizeof(unsigned);
  float*    mf    = (float*)ws;    ws += (size_t)nh * sizeof(float);
  float*    denom = (float*)ws;

  // 1) dense GEMM: TDM-staged LDS operands + fp32 WMMA
  gat_gemm_wmma<<<dim3((n + 15) / 16, 2), dim3(32, 4), 0, stream>>>(x, W, h, n);
  // 2) attention halves
  gat_alpha<<<(nh + 255) / 256, 256, 0, stream>>>(h, a_src, a_dst, as, ad, nh);
  // 3) segment-max: seed with self loops, then edge atomic max
  gat_init_max<<<(nh + 255) / 256, 256, 0, stream>>>(as, ad, menc, nh);
  gat_edge_max<<<(E * HEADS + 255) / 256, 256, 0, stream>>>(ei, as, ad, menc, E);
  // 4) softmax denominator: seed with self loops, then edge atomic add
  gat_init_denom<<<(nh + 255) / 256, 256, 0, stream>>>(as, ad, menc, mf, denom, nh);
  gat_edge_denom<<<(E * HEADS + 255) / 256, 256, 0, stream>>>(ei, as, ad, mf, denom, E);
  // 5) out = bias + self contribution, then per-edge weighted scatter
  gat_init_out<<<(n * HC + 255) / 256, 256, 0, stream>>>(h, as, ad, mf, denom, bias, out, n);
  gat_edge_aggr<<<(int)(((size_t)E * 32 + 255) / 256), 256, 0, stream>>>(
      ei, h, as, ad, mf, denom, out, E);
}